// PyGCN_90512140796730
// MI455X (gfx1250) — compile-verified
//
#include <hip/hip_runtime.h>

#define NN 100000
#define NE 1600000
#define D  64

typedef float v2f __attribute__((ext_vector_type(2)));
typedef float v8f __attribute__((ext_vector_type(8)));

// ---------------------------------------------------------------- utilities
__global__ void zero_f32(float* __restrict__ p, int n) {
    int t = blockIdx.x * blockDim.x + threadIdx.x;
    if (t < n) p[t] = 0.0f;
}

// deg[col[e]] += w[e]  (edge part of segment_sum for degree)
__global__ void edge_deg(const int* __restrict__ col, const float* __restrict__ w,
                         float* __restrict__ deg, int nE) {
    int e = blockIdx.x * blockDim.x + threadIdx.x;
    if (e < nE) unsafeAtomicAdd(&deg[col[e]], w[e]);
}

// deg += 1 (self loop), then dinv = deg>0 ? rsqrt(max(deg,1e-12)) : 0 (in place)
__global__ void node_dinv(float* __restrict__ deg, int n) {
    int i = blockIdx.x * blockDim.x + threadIdx.x;
    if (i < n) {
        float d = deg[i] + 1.0f;
        deg[i] = (d > 0.0f) ? rsqrtf(fmaxf(d, 1e-12f)) : 0.0f;
    }
}

// ---------------------------------------------------------------- GEMM: H = X @ W
// One block = 16 output rows (M tile), 4 waves each owning one 16-wide N tile.
// K=64 consumed as 16x V_WMMA_F32_16X16X4_F32.
__global__ __launch_bounds__(128) void gemm_wmma_f32(const float* __restrict__ X,
                                                     const float* __restrict__ W,
                                                     float* __restrict__ H) {
    __shared__ float WT[D * 68];   // W transposed: WT[n*68 + k] = W[k*D + n]; pad stride 68
    const int tid = threadIdx.x;
    for (int idx = tid; idx < D * D; idx += 128) {
        int k = idx >> 6, n = idx & 63;
        WT[n * 68 + k] = W[idx];
    }
    __syncthreads();

    const int wave  = tid >> 5;            // 0..3 -> N tile
    const int lane  = tid & 31;
    const int l15   = lane & 15;
    const int kh    = (lane >> 4) << 1;    // K sub-offset: 0 (lanes 0-15) or 2 (lanes 16-31)
    const int mbase = blockIdx.x * 16;     // 6250 blocks, N_NODES % 16 == 0
    const int nbase = wave * 16;

    const float* arow = X + (size_t)(mbase + l15) * D + kh;   // A: row M=l15, K=k+kh{,+1}
    const float* bcol = WT + (nbase + l15) * 68 + kh;         // B: col N=l15, K=k+kh{,+1}

    v8f acc = {};
#pragma unroll
    for (int k = 0; k < D; k += 4) {
        v2f a, b;
        a.x = arow[k];     a.y = arow[k + 1];
        b.x = bcol[k];     b.y = bcol[k + 1];
        acc = __builtin_amdgcn_wmma_f32_16x16x4_f32(false, a, false, b,
                                                    (short)0, acc, false, false);
    }

    // D layout: VGPR j -> M = j (+8 for upper half-wave), N = lane&15
    const int row0 = mbase + ((lane >> 4) << 3);
#pragma unroll
    for (int j = 0; j < 8; ++j)
        H[(size_t)(row0 + j) * D + nbase + l15] = acc[j];
}

// ---------------------------------------------------------------- acc[i][:] = b[:]
__global__ void init_acc(float4* __restrict__ acc, const float* __restrict__ b, int nNodes) {
    int t = blockIdx.x * blockDim.x + threadIdx.x;
    if (t < nNodes * (D / 4)) {
        const float4* b4 = (const float4*)b;
        acc[t] = b4[t & 15];
    }
}

// ---------------------------------------------------------------- edge scatter
// One wave per edge: norm = dinv[r]*w*dinv[c]; acc[c][d] += norm*h[r][d], d=lane, lane+32
__global__ __launch_bounds__(256) void edge_scatter(const int* __restrict__ row,
                                                    const int* __restrict__ col,
                                                    const float* __restrict__ w,
                                                    const float* __restrict__ dinv,
                                                    const float* __restrict__ H,
                                                    float* __restrict__ acc, int nE) {
    int e    = blockIdx.x * 8 + (threadIdx.x >> 5);
    int lane = threadIdx.x & 31;
    if (e >= nE) return;
    int r = row[e], c = col[e];
    float nv = dinv[r] * w[e] * dinv[c];
    const float* hr = H + (size_t)r * D;
    float*       ac = acc + (size_t)c * D;
    unsafeAtomicAdd(&ac[lane],      nv * hr[lane]);
    unsafeAtomicAdd(&ac[lane + 32], nv * hr[lane + 32]);
}

// ---------------------------------------------------------------- self-loop + ReLU
// acc = relu(acc + dinv[i]^2 * h)   (acc already holds bias + edge sums)
__global__ void node_finish(float* __restrict__ acc, const float* __restrict__ H,
                            const float* __restrict__ dinv, int nNodes) {
    int t = blockIdx.x * blockDim.x + threadIdx.x;
    if (t < nNodes * D) {
        int i = t >> 6;
        float s = dinv[i]; s = s * s;
        float v = acc[t] + s * H[t];
        acc[t] = v > 0.0f ? v : 0.0f;
    }
}

// ---------------------------------------------------------------- launch
extern "C" void kernel_launch(void* const* d_in, const int* in_sizes, int n_in,
                              void* d_out, int out_size, void* d_ws, size_t ws_size,
                              hipStream_t stream) {
    const float* x   = (const float*)d_in[0];   // [N, D]
    const int*   adj = (const int*)  d_in[1];   // [2, E] (int32 in practice)
    const float* wts = (const float*)d_in[2];   // [E]
    const float* W1  = (const float*)d_in[3];
    const float* b1  = (const float*)d_in[4];
    const float* W2  = (const float*)d_in[5];
    const float* b2  = (const float*)d_in[6];
    float* out = (float*)d_out;

    const int* rowp = adj;        // adj[0]
    const int* colp = adj + NE;   // adj[1]

    char*  ws   = (char*)d_ws;
    float* dinv = (float*)ws;                              // N floats   (0.4 MB)
    float* h    = (float*)(ws + (size_t)NN * sizeof(float));  // N*D floats (25.6 MB)
    float* acc  = h + (size_t)NN * D;                      // N*D floats (25.6 MB)

    const int B = 256;
    // --- normalization ---
    zero_f32 <<<(NN + B - 1) / B, B, 0, stream>>>(dinv, NN);
    edge_deg <<<(NE + B - 1) / B, B, 0, stream>>>(colp, wts, dinv, NE);
    node_dinv<<<(NN + B - 1) / B, B, 0, stream>>>(dinv, NN);

    // --- layer 1 ---
    gemm_wmma_f32<<<NN / 16, 128, 0, stream>>>(x, W1, h);
    init_acc     <<<(NN * 16 + B - 1) / B, B, 0, stream>>>((float4*)acc, b1, NN);
    edge_scatter <<<(NE + 7) / 8, B, 0, stream>>>(rowp, colp, wts, dinv, h, acc, NE);
    node_finish  <<<(NN * D + B - 1) / B, B, 0, stream>>>(acc, h, dinv, NN);  // acc = relu layer1

    // --- layer 2 (accumulate directly into d_out) ---
    gemm_wmma_f32<<<NN / 16, 128, 0, stream>>>(acc, W2, h);
    init_acc     <<<(NN * 16 + B - 1) / B, B, 0, stream>>>((float4*)out, b2, NN);
    edge_scatter <<<(NE + 7) / 8, B, 0, stream>>>(rowp, colp, wts, dinv, h, out, NE);
    node_finish  <<<(NN * D + B - 1) / B, B, 0, stream>>>(out, h, dinv, NN);
}